// KDCR_72885595013359
// MI455X (gfx1250) — compile-verified
//
#include <hip/hip_runtime.h>
#include <math.h>

// ---------------------------------------------------------------------------
// KD loss with teacher-logit rotation, MI455X (gfx1250) version.
//
// Per row r (one 1024-thread workgroup per row):
//   Phase A: async-copy teacher row -> LDS (128KB fits: CDNA5 has 320KB/WGP)
//   Phase B: teacher max M_t, Z_t = sum e^((x-M_t)/T), E_t = sum e^u * u
//            (rotation is a permutation => these are rotation-invariant)
//   Phase C: student max M_s, Z1 = sum e^(x-M_s), ZT = sum e^((x-M_s)/T)
//   Phase D: in-LDS bitonic sort of the teacher row, descending
//   Phase E: cross term: for each j, rotated value is
//              j==label      -> M_t (old max)
//              x_j > x_label -> predecessor value (binary search in sorted LDS)
//              else          -> x_j
//            accumulate e^((rot-M_t)/T) * logp_stu(T)[j]
//   row_out[r] = (1-a)/B * CE_r + a*T^2/(B*C) * KD_r
// Final kernel: reduce the 2048 per-row values with V_WMMA_F32_16X16X4_F32
// (B = ones => D[m,n] = sum_k A[m,k] + C[m,n]; exact f32 reduction).
// ---------------------------------------------------------------------------

#define ALPHA_C   0.9f
#define T_C       4.0f
#define B_ROWS    2048
#define C_COLS    32000
#define C_PAD     32768
#define NTHREADS  1024
#define NWAVES    32

typedef float v2f __attribute__((ext_vector_type(2)));
typedef float v8f __attribute__((ext_vector_type(8)));
typedef int   v4i __attribute__((vector_size(16)));

#if defined(__HIP_DEVICE_COMPILE__)
typedef __attribute__((address_space(1))) v4i* gv4i_p;   // global
typedef __attribute__((address_space(3))) v4i* lv4i_p;   // LDS
#endif

__device__ __forceinline__ float block_reduce(float v, bool is_max, float* red) {
    const int tid  = threadIdx.x;
    const int lane = tid & 31;
    const int wid  = tid >> 5;
    #pragma unroll
    for (int off = 16; off; off >>= 1) {
        float o = __shfl_down(v, off, 32);
        v = is_max ? fmaxf(v, o) : (v + o);
    }
    if (lane == 0) red[wid] = v;
    __syncthreads();
    if (wid == 0) {
        float r = red[lane];   // NWAVES == 32 exactly
        #pragma unroll
        for (int off = 16; off; off >>= 1) {
            float o = __shfl_down(r, off, 32);
            r = is_max ? fmaxf(r, o) : (r + o);
        }
        if (lane == 0) red[0] = r;
    }
    __syncthreads();
    float out = red[0];
    __syncthreads();           // protect red[] for the next call
    return out;
}

__global__ __launch_bounds__(NTHREADS)
void kd_row_kernel(const float* __restrict__ stu,
                   const float* __restrict__ tch,
                   const int*   __restrict__ label,
                   float*       __restrict__ row_out)
{
    __shared__ __align__(16) float s_vals[C_PAD];   // 128 KB: teacher row, then sorted
    __shared__ float red[NWAVES];
    __shared__ float s_scal[4];

    const int row = blockIdx.x;
    const int tid = threadIdx.x;
    const float* trow = tch + (size_t)row * C_COLS;
    const float* srow = stu + (size_t)row * C_COLS;
    const float invT = 1.0f / T_C;

    // ---- Phase A: teacher row -> LDS ------------------------------------
#if defined(__HIP_DEVICE_COMPILE__) && __has_builtin(__builtin_amdgcn_global_load_async_to_lds_b128)
    for (int j4 = tid; j4 < C_COLS / 4; j4 += NTHREADS) {
        __builtin_amdgcn_global_load_async_to_lds_b128(
            (gv4i_p)(trow + 4 * j4),
            (lv4i_p)(s_vals + 4 * j4),
            0, 0);
    }
  #if __has_builtin(__builtin_amdgcn_s_wait_asynccnt)
    __builtin_amdgcn_s_wait_asynccnt(0);
  #else
    asm volatile("s_wait_asynccnt 0" ::: "memory");
  #endif
#else
    for (int j4 = tid; j4 < C_COLS / 4; j4 += NTHREADS) {
        ((float4*)s_vals)[j4] = ((const float4*)trow)[j4];
    }
#endif
    for (int j = C_COLS + tid; j < C_PAD; j += NTHREADS) s_vals[j] = -INFINITY;
    if (tid == 0) {
        int lab = label[row];
        s_scal[0] = trow[lab];               // lv: teacher logit at label
        s_scal[1] = srow[lab];               // student logit at label
        s_scal[2] = __int_as_float(lab);
    }
    __syncthreads();
    const float lv   = s_scal[0];
    const float slab = s_scal[1];
    const int   lab  = __float_as_int(s_scal[2]);

    // ---- Phase B: teacher stats (rotation-invariant) --------------------
    float m = -INFINITY;
    for (int j = tid; j < C_COLS; j += NTHREADS) m = fmaxf(m, s_vals[j]);
    const float Mt = block_reduce(m, true, red);

    float zt = 0.0f, et = 0.0f;
    for (int j = tid; j < C_COLS; j += NTHREADS) {
        float u = (s_vals[j] - Mt) * invT;
        float e = __expf(u);
        zt += e;
        et += e * u;
    }
    const float Zt = block_reduce(zt, false, red);
    const float Et = block_reduce(et, false, red);

    // ---- Phase C: student stats -----------------------------------------
    float ms = -INFINITY;
    for (int j = tid; j < C_COLS; j += NTHREADS) ms = fmaxf(ms, srow[j]);
    const float Ms = block_reduce(ms, true, red);

    float z1 = 0.0f, zT = 0.0f;
    for (int j = tid; j < C_COLS; j += NTHREADS) {
        float d = srow[j] - Ms;
        z1 += __expf(d);
        zT += __expf(d * invT);
    }
    const float Z1   = block_reduce(z1, false, red);
    const float ZT   = block_reduce(zT, false, red);
    const float lnZT = __logf(ZT);

    // ---- Phase D: bitonic sort of teacher row, descending ---------------
    __syncthreads();
    for (int k = 2; k <= C_PAD; k <<= 1) {
        for (int jj = k >> 1; jj > 0; jj >>= 1) {
            for (int i = tid; i < C_PAD; i += NTHREADS) {
                int ixj = i ^ jj;
                if (ixj > i) {
                    float a = s_vals[i];
                    float b = s_vals[ixj];
                    bool upper = ((i & k) == 0);
                    // descending overall: invert the classic ascending compare
                    if (upper ? (a < b) : (a > b)) {
                        s_vals[i]   = b;
                        s_vals[ixj] = a;
                    }
                }
            }
            __syncthreads();
        }
    }

    // ---- Phase E: cross term over rotated teacher -----------------------
    float cross = 0.0f;
    for (int j = tid; j < C_COLS; j += NTHREADS) {
        float x   = trow[j];                       // L2-resident re-read
        float su  = srow[j];
        float lps = (su - Ms) * invT - lnZT;       // log_softmax(stu/T)[j]
        float rotv;
        if (j == lab) {
            rotv = Mt;                             // label slot gets old max
        } else if (x > lv) {
            // binary search for x in descending s_vals (values distinct)
            int lo = 0, hi = C_COLS - 1;
            while (lo < hi) {
                int mid = (lo + hi) >> 1;
                if (s_vals[mid] > x) lo = mid + 1; else hi = mid;
            }
            rotv = s_vals[lo + 1];                 // predecessor value
        } else {
            rotv = x;                              // below label: unchanged
        }
        cross += __expf((rotv - Mt) * invT) * lps;
    }
    const float Cr = block_reduce(cross, false, red);

    if (tid == 0) {
        float invZt  = 1.0f / Zt;
        float kd_row = (Et * invZt - __logf(Zt)) - Cr * invZt;   // sum_j p(logp_t - logp_s)
        float ce_row = -(slab - Ms - __logf(Z1));
        row_out[row] = (1.0f - ALPHA_C) / (float)B_ROWS * ce_row
                     + ALPHA_C * T_C * T_C / ((float)B_ROWS * (float)C_COLS) * kd_row;
    }
}

// ---- Final reduction of 2048 per-row values via WMMA ----------------------
// D = A x Ones + C  (16x16x4 f32): D[m,n] = sum_k A[m,k] + C[m,n].
// With ones in B the mapping of data to (m,k) slots is irrelevant for the
// grand total: total = sum_m D[m,0] = (sum of 8 D VGPRs in lane 0)
//                                   + (sum of 8 D VGPRs in lane 16).
__global__ __launch_bounds__(32)
void final_reduce_kernel(const float* __restrict__ row_vals,
                         float* __restrict__ out)
{
    const int lane = threadIdx.x;
#if defined(__HIP_DEVICE_COMPILE__) && __has_builtin(__builtin_amdgcn_wmma_f32_16x16x4_f32)
    v8f acc = {};
    v2f ones;
    ones[0] = 1.0f; ones[1] = 1.0f;
    #pragma unroll 4
    for (int it = 0; it < B_ROWS / 64; ++it) {
        v2f a;
        a[0] = row_vals[it * 64 + lane];
        a[1] = row_vals[it * 64 + 32 + lane];
        acc = __builtin_amdgcn_wmma_f32_16x16x4_f32(
            false, a, false, ones, (short)0, acc, false, false);
    }
    float L = acc[0] + acc[1] + acc[2] + acc[3] + acc[4] + acc[5] + acc[6] + acc[7];
    float other = __shfl(L, lane ^ 16, 32);
    if (lane == 0) out[0] = L + other;
#else
    float s = 0.0f;
    for (int i = lane; i < B_ROWS; i += 32) s += row_vals[i];
    #pragma unroll
    for (int off = 16; off; off >>= 1) s += __shfl_down(s, off, 32);
    if (lane == 0) out[0] = s;
#endif
}

extern "C" void kernel_launch(void* const* d_in, const int* in_sizes, int n_in,
                              void* d_out, int out_size, void* d_ws, size_t ws_size,
                              hipStream_t stream) {
    (void)in_sizes; (void)n_in; (void)out_size; (void)ws_size;
    const float* stu   = (const float*)d_in[0];
    const float* tch   = (const float*)d_in[1];
    const int*   label = (const int*)d_in[2];
    float* out    = (float*)d_out;
    float* row_ws = (float*)d_ws;          // 2048 floats of scratch

    kd_row_kernel<<<B_ROWS, NTHREADS, 0, stream>>>(stu, tch, label, row_ws);
    final_reduce_kernel<<<1, 32, 0, stream>>>(row_ws, out);
}